// FrameEnergyLoss_12146167513822
// MI455X (gfx1250) — compile-verified
//
#include <hip/hip_runtime.h>
#include <hip/hip_bf16.h>

typedef float v2f __attribute__((ext_vector_type(2)));
typedef float v8f __attribute__((ext_vector_type(8)));

#define NBLOCKS 1024
#define NTHREADS 256

// Phase 1: fused element-energy + external-work partial reduction.
// Each block writes one float partial (energy minus work) to d_ws.
__global__ __launch_bounds__(NTHREADS) void frame_energy_partial(
    const float* __restrict__ pred_raw,      // (n_nodes,3)
    const float* __restrict__ elem_len,      // (n_elem)
    const float* __restrict__ prop_E,        // (n_elem)
    const float* __restrict__ prop_A,        // (n_elem)
    const float* __restrict__ prop_I22,      // (n_elem)
    const float* __restrict__ elem_dir,      // (n_elem,3): c = [3e], s = [3e+2]
    const float* __restrict__ F_ext,         // (n_nodes,3)
    const float* __restrict__ u_c_p,
    const float* __restrict__ theta_c_p,
    const int*   __restrict__ conn,          // (n_elem,2)
    float*       __restrict__ partials,      // (NBLOCKS)
    int n_elem, int n_nodes)
{
    const float u_c  = u_c_p[0];
    const float th_c = theta_c_p[0];

    const int tid    = blockIdx.x * blockDim.x + threadIdx.x;
    const int stride = gridDim.x * blockDim.x;

    float acc = 0.0f;

    // ---- Element strain energy (analytic 6x6 quadratic form) ----
    for (int e = tid; e < n_elem; e += stride) {
        const int nA = conn[2 * e + 0];
        const int nB = conn[2 * e + 1];

        const float L    = elem_len[e];
        const float Ee   = prop_E[e];
        const float EA   = Ee * prop_A[e];
        const float EI   = Ee * prop_I22[e];
        const float c    = elem_dir[3 * e + 0];
        const float s    = elem_dir[3 * e + 2];

        const float ax = pred_raw[3 * nA + 0] * u_c;
        const float ay = pred_raw[3 * nA + 1] * u_c;
        const float tA = pred_raw[3 * nA + 2] * th_c;
        const float bx = pred_raw[3 * nB + 0] * u_c;
        const float by = pred_raw[3 * nB + 1] * u_c;
        const float tB = pred_raw[3 * nB + 2] * th_c;

        const float uA =  c * ax + s * ay;
        const float wA = -s * ax + c * ay;
        const float uB =  c * bx + s * by;
        const float wB = -s * bx + c * by;

        const float invL  = 1.0f / L;
        const float ea_L  = EA * invL;
        const float ei_L  = EI * invL;
        const float ei_L2 = ei_L * invL;
        const float ei_L3 = ei_L2 * invL;

        const float du = uA - uB;
        const float dw = wA - wB;
        const float ts = tA + tB;

        const float quad = ea_L * du * du
                         + 12.0f * ei_L3 * dw * dw
                         + 12.0f * ei_L2 * dw * ts
                         +  4.0f * ei_L  * (tA * tA + tA * tB + tB * tB);
        acc += 0.5f * quad;
    }

    // ---- External work: subtract F_ext . u_phys ----
    for (int n = tid; n < n_nodes; n += stride) {
        const float w = F_ext[3 * n + 0] * (pred_raw[3 * n + 0] * u_c)
                      + F_ext[3 * n + 1] * (pred_raw[3 * n + 1] * u_c)
                      + F_ext[3 * n + 2] * (pred_raw[3 * n + 2] * th_c);
        acc -= w;
    }

    // ---- Block tree reduction in LDS ----
    __shared__ float sdata[NTHREADS];
    sdata[threadIdx.x] = acc;
    __syncthreads();
    for (int off = NTHREADS / 2; off > 0; off >>= 1) {
        if (threadIdx.x < off) sdata[threadIdx.x] += sdata[threadIdx.x + off];
        __syncthreads();
    }
    if (threadIdx.x == 0) partials[blockIdx.x] = sdata[0];
}

// Phase 2: one wave32 reduces NBLOCKS partials. Final cross-lane sum is done
// on the matrix pipe: A(16x4, f32) x ones(4x16) via V_WMMA_F32_16X16X4_F32.
// A-layout (ISA 7.12.2): VGPR0 lanes 0-15 = A[m][0], lanes 16-31 = A[m][2];
// VGPR1 = A[m][1], A[m][3] (set to 0). With B = all-ones:
//   D[m][n] = partial_lane_m + partial_lane_(m+16)   for every n.
// Column 0 lives in lane 0 (rows 0-7, D VGPRs 0-7) and lane 16 (rows 8-15).
__global__ __launch_bounds__(32) void frame_energy_finalize(
    const float* __restrict__ partials,
    const float* __restrict__ u_c_p,
    const float* __restrict__ F_c_p,
    float*       __restrict__ out,
    int nparts)
{
    const int lane = threadIdx.x;   // exactly one wave32, EXEC all ones

    float s = 0.0f;
    for (int i = lane; i < nparts; i += 32) s += partials[i];

    v2f a;  a[0] = s;    a[1] = 0.0f;   // A: K=0/K=2 carry partials, K=1/K=3 zero
    v2f b;  b[0] = 1.0f; b[1] = 1.0f;   // B: all ones (layout-independent)
    v8f cz = {};

    v8f d = __builtin_amdgcn_wmma_f32_16x16x4_f32(
        /*neg_a=*/false, a, /*neg_b=*/false, b,
        /*c_mod=*/(short)0, cz, /*reuse_a=*/false, /*reuse_b=*/false);

    // Per-lane sum of its 8 rows at column (lane % 16).
    float rowsum = d[0] + d[1] + d[2] + d[3] + d[4] + d[5] + d[6] + d[7];

    __shared__ float sm[32];
    sm[lane] = rowsum;
    __syncthreads();

    if (lane == 0) {
        const float total = sm[0] + sm[16];          // column 0: rows 0-7 + rows 8-15
        const float E_c   = fmaxf(F_c_p[0] * u_c_p[0], 1e-30f);
        out[0] = total / E_c;
    }
}

extern "C" void kernel_launch(void* const* d_in, const int* in_sizes, int n_in,
                              void* d_out, int out_size, void* d_ws, size_t ws_size,
                              hipStream_t stream) {
    const float* pred_raw  = (const float*)d_in[0];
    const float* elem_len  = (const float*)d_in[1];
    const float* prop_E    = (const float*)d_in[2];
    const float* prop_A    = (const float*)d_in[3];
    const float* prop_I22  = (const float*)d_in[4];
    const float* elem_dir  = (const float*)d_in[5];
    const float* F_ext     = (const float*)d_in[6];
    const float* u_c       = (const float*)d_in[7];
    const float* theta_c   = (const float*)d_in[8];
    const float* F_c       = (const float*)d_in[9];
    const int*   conn      = (const int*)d_in[10];

    const int n_elem  = in_sizes[1];
    const int n_nodes = in_sizes[0] / 3;

    float* partials = (float*)d_ws;       // NBLOCKS floats of scratch
    float* out      = (float*)d_out;

    frame_energy_partial<<<NBLOCKS, NTHREADS, 0, stream>>>(
        pred_raw, elem_len, prop_E, prop_A, prop_I22, elem_dir, F_ext,
        u_c, theta_c, conn, partials, n_elem, n_nodes);

    frame_energy_finalize<<<1, 32, 0, stream>>>(partials, u_c, F_c, out, NBLOCKS);
}